// HistogramLoss_45964740002307
// MI455X (gfx1250) — compile-verified
//
#include <hip/hip_runtime.h>
#include <hip/hip_bf16.h>

// ---------------- problem constants ----------------
#define BATCH   16
#define CHAN    3
#define HDIM    512
#define WDIM    512
#define HW      (HDIM * WDIM)          // 262144
#define NIDX    131072                 // gather count per batch
#define NBINS   256
#define TOTAL_ELEMS (BATCH * CHAN * HW) // 12,582,912

// loss kernel tiling: 64 blocks/plane, each covers 4096 elements
#define LBLK_X  64
#define TILE    1024                   // elements per pipeline tile (256 thr * 4)
#define NTILES  4                      // 4 tiles * 1024 = 4096 per block
#define LITER   16                     // scalar fallback: 16 * 256 = 4096
#define NPART   (BATCH * CHAN * LBLK_X) // 3072 fp64 partials

typedef float v2f __attribute__((ext_vector_type(2)));
typedef float v8f __attribute__((ext_vector_type(8)));
typedef int   v4i __attribute__((ext_vector_type(4)));

// ---------------- CDNA5 async-to-LDS availability ----------------
#if defined(__AMDGCN__)
#  if __has_builtin(__builtin_amdgcn_global_load_async_to_lds_b32) && \
      __has_builtin(__builtin_amdgcn_global_load_async_to_lds_b8)  && \
      __has_builtin(__builtin_amdgcn_s_wait_asynccnt)
#    define ASYNC_OK 1
#  else
#    define ASYNC_OK 0
#    warning "gfx1250 async-to-LDS builtins not found; loss kernel using direct-load fallback"
#  endif
#  if ASYNC_OK && __has_builtin(__builtin_amdgcn_global_load_async_to_lds_b128)
#    define ASYNC128 1
#  else
#    define ASYNC128 0
#  endif
#else
#  define ASYNC_OK  0
#  define ASYNC128  0
#endif

#if ASYNC_OK
#define AS1 __attribute__((address_space(1)))
#define AS3 __attribute__((address_space(3)))
__device__ __forceinline__ void async_cp_b32(const void* g, void* l) {
    __builtin_amdgcn_global_load_async_to_lds_b32(
        (AS1 int*)g, (AS3 int*)l, /*offset=*/0, /*cpol=*/0);
}
__device__ __forceinline__ void async_cp_b8(const void* g, void* l) {
    __builtin_amdgcn_global_load_async_to_lds_b8(
        (AS1 char*)g, (AS3 char*)l, /*offset=*/0, /*cpol=*/0);
}
#endif
#if ASYNC128
__device__ __forceinline__ void async_cp_b128(const void* g, void* l) {
    __builtin_amdgcn_global_load_async_to_lds_b128(
        (AS1 v4i*)g, (AS3 v4i*)l, /*offset=*/0, /*cpol=*/0);
}
#endif

__device__ __forceinline__ float denorm(float v) {
    // clip((v+1)*0.5, 0, 1) * 255
    return fminf(fmaxf((v + 1.0f) * 0.5f, 0.0f), 1.0f) * 255.0f;
}

// ---------------- kernel 0: zero scratch ----------------
__global__ void zero_kernel(unsigned* __restrict__ p, int nwords) {
    int i = blockIdx.x * blockDim.x + threadIdx.x;
    int stride = gridDim.x * blockDim.x;
    for (; i < nwords; i += stride) p[i] = 0u;
}

// ---------------- kernel 1: histograms + scatter flags ----------------
// grid (BATCH, 64 chunks), 256 threads. LDS: 6 histograms of 256 u32 = 6KB.
__global__ void hist_kernel(const float* __restrict__ y,
                            const float* __restrict__ ref,
                            const float* __restrict__ msrc,
                            const float* __restrict__ mtgt,
                            const long long* __restrict__ idx,
                            unsigned* __restrict__ histD,
                            unsigned* __restrict__ histR,
                            unsigned char* __restrict__ flags) {
    __shared__ unsigned hD[CHAN][NBINS];
    __shared__ unsigned hR[CHAN][NBINS];
    const int t = threadIdx.x;
    const int b = blockIdx.x;
    for (int i = t; i < CHAN * NBINS; i += 256) {
        (&hD[0][0])[i] = 0u;
        (&hR[0][0])[i] = 0u;
    }
    __syncthreads();

    const long long* I = idx + (size_t)b * 4 * NIDX;
    const int n0 = blockIdx.y * 2048;
    for (int k = 0; k < 8; ++k) {
        const int n  = n0 + k * 256 + t;
        const int i0 = (int)I[(size_t)0 * NIDX + n];
        const int i1 = (int)I[(size_t)1 * NIDX + n];
        const int i2 = (int)I[(size_t)2 * NIDX + n];
        const int i3 = (int)I[(size_t)3 * NIDX + n];
        const int pd = i0 * WDIM + i1;   // scatter/dst position
        const int pr = i2 * WDIM + i3;   // reference-gather position
        const float ms = msrc[(size_t)b * HW + pd];
        const float mt = mtgt[(size_t)b * HW + pr];
        flags[(size_t)b * HW + pd] = (unsigned char)1;  // duplicates write same value
#pragma unroll
        for (int c = 0; c < CHAN; ++c) {
            const size_t plane = (size_t)(b * CHAN + c) * HW;
            const float vd = denorm(ref[plane + pd]) * ms;
            int bd = (int)vd; bd = min(max(bd, 0), NBINS - 1);
            atomicAdd(&hD[c][bd], 1u);
            const float vr = denorm(y[plane + pr]) * mt;
            int br = (int)vr; br = min(max(br, 0), NBINS - 1);
            atomicAdd(&hR[c][br], 1u);
        }
    }
    __syncthreads();
    for (int i = t; i < CHAN * NBINS; i += 256) {
        const int c = i >> 8, bin = i & 255;
        const unsigned vD = hD[c][bin];
        if (vD) atomicAdd(&histD[(size_t)(b * CHAN + c) * NBINS + bin], vD);
        const unsigned vR = hR[c][bin];
        if (vR) atomicAdd(&histR[(size_t)(b * CHAN + c) * NBINS + bin], vR);
    }
}

// ---------------- kernel 2: cumsum -> CDF -> matching table ----------------
// grid 48 blocks (one per b,c), 256 threads. Exact fp32 (counts <= 2^17).
__global__ void cdf_table_kernel(const unsigned* __restrict__ histD,
                                 const unsigned* __restrict__ histR,
                                 float* __restrict__ tablef) {
    __shared__ float cD[NBINS];
    __shared__ float cR[NBINS];
    const int bc = blockIdx.x;
    const int t  = threadIdx.x;
    cD[t] = (float)histD[(size_t)bc * NBINS + t];
    cR[t] = (float)histR[(size_t)bc * NBINS + t];
    __syncthreads();
    // Hillis-Steele inclusive scan (exact: integer sums <= 2^17 < 2^24)
    for (int off = 1; off < NBINS; off <<= 1) {
        const float aD = (t >= off) ? cD[t - off] : 0.0f;
        const float aR = (t >= off) ? cR[t - off] : 0.0f;
        __syncthreads();
        cD[t] += aD;
        cR[t] += aR;
        __syncthreads();
    }
    const float inv = 1.0f / (float)NIDX;   // totals are exactly NIDX
    const float r = cD[t] * inv;
    int tab = t;
    for (int j = 0; j < NBINS - 1; ++j) {
        const float lo = cR[j] * inv;
        const float hi = cR[j + 1] * inv;
        if (lo <= r && r <= hi) { tab = j + 1; break; }   // first j (argmax) + 1
    }
    if (t == NBINS - 1) tab = NBINS - 1;
    tablef[(size_t)bc * NBINS + t] = (float)tab;
}

// ---------------- kernel 3: streaming loss ----------------
// Double-buffered async-to-LDS pipeline (ASYNCcnt, b128 width) + WMMA wave
// reduction. grid (64, CHAN, BATCH), 256 threads; no divergence -> EXEC all-1s.
// Each lane's async ops fill exactly the LDS slots that same lane consumes,
// so the pipeline needs only per-wave s_wait_asynccnt (no barriers).
__global__ void loss_kernel(const float* __restrict__ x,
                            const float* __restrict__ ref,
                            const float* __restrict__ msrc,
                            const unsigned char* __restrict__ flags,
                            const float* __restrict__ tablef,
                            double* __restrict__ partials) {
    __shared__ float sTab[NBINS];
    __shared__ float wsum[8];
#if ASYNC128
    __shared__ float sX [2][TILE];
    __shared__ float sRf[2][TILE];
    __shared__ float sM [2][TILE];
    __shared__ unsigned char sF[2][TILE];
#elif ASYNC_OK
    __shared__ float sX [2][256];
    __shared__ float sRf[2][256];
    __shared__ float sM [2][256];
    __shared__ unsigned char sF[2][256];
#endif
    const int t = threadIdx.x;
    const int b = blockIdx.z;
    const int c = blockIdx.y;
    const size_t plane = (size_t)(b * CHAN + c) * HW;
    const size_t mbase = (size_t)b * HW;
    sTab[t] = tablef[(size_t)(b * CHAN + c) * NBINS + t];
    __syncthreads();

    float acc = 0.0f;

#if ASYNC128
    // 4 tiles of 1024 elements; each lane moves 16B (b128) per array per tile.
    const int p0 = blockIdx.x * (TILE * NTILES);
    const int q  = 4 * t;                      // this lane's slot in the tile
    {   // prime tile 0 into buffer 0
        const int p = p0 + q;
        async_cp_b128(&x  [plane + p], &sX [0][q]);
        async_cp_b128(&ref[plane + p], &sRf[0][q]);
        async_cp_b128(&msrc[mbase + p], &sM [0][q]);
        async_cp_b32 (&flags[mbase + p], &sF [0][q]);
    }
#pragma unroll
    for (int k = 0; k < NTILES; ++k) {
        const int buf = k & 1;
        if (k + 1 < NTILES) {
            const int pn = p0 + (k + 1) * TILE + q;
            async_cp_b128(&x  [plane + pn], &sX [buf ^ 1][q]);
            async_cp_b128(&ref[plane + pn], &sRf[buf ^ 1][q]);
            async_cp_b128(&msrc[mbase + pn], &sM [buf ^ 1][q]);
            async_cp_b32 (&flags[mbase + pn], &sF [buf ^ 1][q]);
            __builtin_amdgcn_s_wait_asynccnt(4);   // tile k resident, k+1 in flight
        } else {
            __builtin_amdgcn_s_wait_asynccnt(0);   // drain
        }
        const float4  xv = *(const float4*)&sX [buf][q];
        const float4  rv = *(const float4*)&sRf[buf][q];
        const float4  mv = *(const float4*)&sM [buf][q];
        const unsigned fw = *(const unsigned*)&sF[buf][q];
#pragma unroll
        for (int j = 0; j < 4; ++j) {
            const float m  = (&mv.x)[j];
            const float im = denorm((&xv.x)[j]) * m;
            const float rm = denorm((&rv.x)[j]) * m;
            int bin = (int)rm; bin = min(max(bin, 0), NBINS - 1);
            const float mt = ((fw >> (8 * j)) & 0xFF) ? sTab[bin] : rm;
            acc += fabsf(im - mt);
        }
    }
#elif ASYNC_OK
    const int p0 = blockIdx.x * (256 * LITER);
    {   // prime tile 0 into buffer 0
        const int p = p0 + t;
        async_cp_b32(&x  [plane + p],  &sX [0][t]);
        async_cp_b32(&ref[plane + p],  &sRf[0][t]);
        async_cp_b32(&msrc[mbase + p], &sM [0][t]);
        async_cp_b8 (&flags[mbase + p], &sF [0][t]);
    }
#pragma unroll
    for (int k = 0; k < LITER; ++k) {
        const int buf = k & 1;
        if (k + 1 < LITER) {
            const int pn = p0 + (k + 1) * 256 + t;
            async_cp_b32(&x  [plane + pn],  &sX [buf ^ 1][t]);
            async_cp_b32(&ref[plane + pn],  &sRf[buf ^ 1][t]);
            async_cp_b32(&msrc[mbase + pn], &sM [buf ^ 1][t]);
            async_cp_b8 (&flags[mbase + pn], &sF [buf ^ 1][t]);
            __builtin_amdgcn_s_wait_asynccnt(4);
        } else {
            __builtin_amdgcn_s_wait_asynccnt(0);
        }
        const float m  = sM[buf][t];
        const float im = denorm(sX [buf][t]) * m;
        const float rm = denorm(sRf[buf][t]) * m;
        int bin = (int)rm; bin = min(max(bin, 0), NBINS - 1);
        const float mt = sF[buf][t] ? sTab[bin] : rm;
        acc += fabsf(im - mt);
    }
#else
    const int p0 = blockIdx.x * (256 * LITER);
#pragma unroll
    for (int k = 0; k < LITER; ++k) {
        const int p = p0 + k * 256 + t;          // always < HW (exact tiling)
        const float m  = msrc[mbase + p];
        const float im = denorm(x[plane + p]) * m;
        const float rm = denorm(ref[plane + p]) * m;
        int bin = (int)rm; bin = min(max(bin, 0), NBINS - 1);
        const float mt = flags[mbase + p] ? sTab[bin] : rm;
        acc += fabsf(im - mt);
    }
#endif

    // ---- wave(32) reduction ----
    float s;
#if defined(__gfx1250__) && __has_builtin(__builtin_amdgcn_wmma_f32_16x16x4_f32)
    // A = all-ones 16x4 (layout-independent), B holds the 32 lane accumulators
    // (+32 zeros) in its 64 slots. E = A*B gives B-column sums; summing one
    // result VGPR across all 32 lanes counts every column sum twice, so the
    // wave total = lane-sum(E[0]) * 0.5. Exact in f32, robust to B layout.
    v2f a  = {1.0f, 1.0f};
    v2f bb = {acc, 0.0f};
    v8f cz = {};
    v8f e = __builtin_amdgcn_wmma_f32_16x16x4_f32(
        /*neg_a=*/false, a, /*neg_b=*/false, bb,
        /*c_mod=*/(short)0, cz, /*reuse_a=*/false, /*reuse_b=*/false);
    s = e[0];
#pragma unroll
    for (int off = 16; off > 0; off >>= 1) s += __shfl_xor(s, off, 32);
    s *= 0.5f;
#else
    s = acc;
#pragma unroll
    for (int off = 16; off > 0; off >>= 1) s += __shfl_xor(s, off, 32);
#endif

    if ((t & 31) == 0) wsum[t >> 5] = s;
    __syncthreads();
    if (t == 0) {
        float bsum = 0.0f;
#pragma unroll
        for (int w = 0; w < 8; ++w) bsum += wsum[w];
        partials[(size_t)(b * CHAN + c) * LBLK_X + blockIdx.x] = (double)bsum;
    }
}

// ---------------- kernel 4: deterministic final reduction ----------------
__global__ void final_kernel(const double* __restrict__ partials,
                             float* __restrict__ out) {
    __shared__ double sd[256];
    const int t = threadIdx.x;
    double s = 0.0;
    for (int i = t; i < NPART; i += 256) s += partials[i];   // fixed order
    sd[t] = s;
    __syncthreads();
    for (int off = 128; off > 0; off >>= 1) {
        if (t < off) sd[t] += sd[t + off];
        __syncthreads();
    }
    if (t == 0) out[0] = (float)(sd[0] / (double)TOTAL_ELEMS);
}

// ---------------- host launch ----------------
extern "C" void kernel_launch(void* const* d_in, const int* in_sizes, int n_in,
                              void* d_out, int out_size, void* d_ws, size_t ws_size,
                              hipStream_t stream) {
    const float*      x    = (const float*)d_in[0];
    const float*      y    = (const float*)d_in[1];
    const float*      msrc = (const float*)d_in[2];
    const float*      mtgt = (const float*)d_in[3];
    const long long*  idx  = (const long long*)d_in[4];   // int64 indices
    const float*      ref  = (const float*)d_in[5];

    // workspace layout (bytes):
    //   [0,        24576)  : 3072 fp64 block partials (fully overwritten)
    //   [24576,    73728)  : histD  48*256 u32
    //   [73728,   122880)  : histR  48*256 u32
    //   [122880,  172032)  : tablef 48*256 f32 (fully overwritten)
    //   [172032, 4366336)  : flags  16*262144 u8
    char* ws = (char*)d_ws;
    double*        partials = (double*)ws;
    unsigned*      histD    = (unsigned*)(ws + 24576);
    unsigned*      histR    = (unsigned*)(ws + 73728);
    float*         tablef   = (float*)(ws + 122880);
    unsigned char* flags    = (unsigned char*)(ws + 172032);

    // zero hists + tablef + flags region each call (determinism)
    const int zwords = (49152 * 3 + BATCH * HW) / 4;   // 1,085,440 words
    zero_kernel<<<2048, 256, 0, stream>>>((unsigned*)(ws + 24576), zwords);

    hist_kernel<<<dim3(BATCH, 64), 256, 0, stream>>>(y, ref, msrc, mtgt, idx,
                                                     histD, histR, flags);
    cdf_table_kernel<<<BATCH * CHAN, 256, 0, stream>>>(histD, histR, tablef);
    loss_kernel<<<dim3(LBLK_X, CHAN, BATCH), 256, 0, stream>>>(x, ref, msrc,
                                                               flags, tablef,
                                                               partials);
    final_kernel<<<1, 256, 0, stream>>>(partials, (float*)d_out);
}